// SparseConv3d_InPlace_Autograd_88373247082542
// MI455X (gfx1250) — compile-verified
//
#include <hip/hip_runtime.h>

typedef __attribute__((ext_vector_type(2))) float v2f;
typedef __attribute__((ext_vector_type(8))) float v8f;

#define K3    27
#define C_IN  32
#define C_OUT 32

// Non-returning fp32 atomic add, device scope -> global_atomic_add_f32 scope:SCOPE_DEV.
__device__ inline void atomic_add_f32(float* p, float v) {
    __hip_atomic_fetch_add(p, v, __ATOMIC_RELAXED, __HIP_MEMORY_SCOPE_AGENT);
}

// Fill output with broadcast bias (d_out is poisoned before timing).
__global__ __launch_bounds__(256) void init_bias_kernel(
    float* __restrict__ out, const float* __restrict__ bias, long long total4) {
    long long i = (long long)blockIdx.x * blockDim.x + threadIdx.x;
    if (i >= total4) return;
    int c4 = (int)(i & (C_OUT / 4 - 1));           // 8 float4 per output row
    ((float4*)out)[i] = ((const float4*)bias)[c4];
}

// Hot kernel: persistent waves. Each wave32 loops over complete 16-row groups
// of kernel-offset blockIdx.y: gather A (16x32 fp32), WMMA against the 32x32
// weight (B fragments held in registers across the loop), scatter-add 16x32.
// No per-row predicates: EXEC stays full from gather to scatter. All global
// accesses use unsigned 32-bit offsets so LLVM selects SADDR addressing
// (SGPR base + VGPR32 offset) instead of 64-bit VALU address math.
__global__ __launch_bounds__(256) void sparse_conv_gather_mm_scatter(
    const float* __restrict__ x_data,    // [N_IN, 32]
    const float* __restrict__ weights,   // [27, 32, 32]
    const int*   __restrict__ in_inds,   // [27, R]
    const int*   __restrict__ out_inds,  // [27, R]
    float*       __restrict__ out,       // [N_OUT, 32]
    int R) {
    __shared__ float lds_w[C_IN * C_OUT];

    const int k = blockIdx.y;
    // Stage this offset's 32x32 weight into LDS, coalesced (256 thr * float4).
    ((float4*)lds_w)[threadIdx.x] =
        ((const float4*)(weights + (size_t)k * C_IN * C_OUT))[threadIdx.x];
    __syncthreads();

    const int lane = threadIdx.x & 31;
    const int wave = threadIdx.x >> 5;
    const int lo   = lane & 15;   // N / M position within half-wave
    const int hi   = lane >> 4;   // half-wave select

    // B fragments (4x16 KxN per WMMA): VGPR0 = B[K=hi][N=lo], VGPR1 = B[K=2+hi][N=lo].
    // Loaded once, live in registers for the whole persistent loop.
    v2f bf[2][8];
#pragma unroll
    for (int t = 0; t < 2; ++t)
#pragma unroll
        for (int kk = 0; kk < 8; ++kk) {
            bf[t][kk][0] = lds_w[(4 * kk + hi)     * C_OUT + 16 * t + lo];
            bf[t][kk][1] = lds_w[(4 * kk + 2 + hi) * C_OUT + 16 * t + lo];
        }

    const long long koff        = (long long)k * R;
    const int       stride_rows = (int)gridDim.x * 8 * 16;   // all waves in x-dim

    for (int r0 = ((int)blockIdx.x * 8 + wave) * 16; r0 + 16 <= R; r0 += stride_rows) {
        const long long base = koff + r0;
        const unsigned ii   = (unsigned)in_inds[base + lo];   // gather row (row = lo)
        const unsigned oidx = (unsigned)out_inds[base + lo];  // scatter row (row = lo)

        // A fragments (16x4 MxK): lane(lo,hi) VGPR0/1 = A[lo][4kk+2hi .. +1].
        // 32-bit element offset -> SADDR-form global_load_b64.
        const float2* xrow = (const float2*)(x_data + (size_t)(ii * (unsigned)C_IN));
        v2f af[8];
#pragma unroll
        for (int kk = 0; kk < 8; ++kk) {
            float2 v = xrow[2 * kk + hi];
            af[kk][0] = v.x;
            af[kk][1] = v.y;
        }

        // [16x32] @ [32x32] = two 16x16 fp32 tiles: 16 x V_WMMA_F32_16X16X4_F32.
        v8f c0 = {}, c1 = {};
#pragma unroll
        for (int kk = 0; kk < 8; ++kk) {
            c0 = __builtin_amdgcn_wmma_f32_16x16x4_f32(false, af[kk], false, bf[0][kk],
                                                       (short)0, c0, false, false);
            c1 = __builtin_amdgcn_wmma_f32_16x16x4_f32(false, af[kk], false, bf[1][kk],
                                                       (short)0, c1, false, false);
        }

        // C/D layout: VGPR j holds row M = j + 8*hi, col N = lo (t=0) / 16+lo (t=1).
        // Hoisted bpermutes (overlapped DS latency), then straight-line atomics
        // with 32-bit unsigned offsets (SADDR form, no 64-bit adds).
        unsigned o[8];
#pragma unroll
        for (int j = 0; j < 8; ++j)
            o[j] = (unsigned)__shfl((int)oidx, j + 8 * hi, 32);
#pragma unroll
        for (int j = 0; j < 8; ++j) {
            const unsigned off = o[j] * (unsigned)C_OUT + (unsigned)lo;
            atomic_add_f32(out + (size_t)off,        c0[j]);
            atomic_add_f32(out + (size_t)(off + 16), c1[j]);
        }
    }
}

// Tail kernel: handles the last (R % 16) rulebook rows of each offset with
// plain per-thread dot products. Launched only when R % 16 != 0.
__global__ __launch_bounds__(32) void sparse_conv_tail(
    const float* __restrict__ x_data,
    const float* __restrict__ weights,
    const int*   __restrict__ in_inds,
    const int*   __restrict__ out_inds,
    float*       __restrict__ out,
    int R, int r_tail) {
    const int k = blockIdx.x;          // one block (one wave) per kernel offset
    const int c = threadIdx.x;         // output channel 0..31
    for (int r = r_tail; r < R; ++r) {
        const long long base = (long long)k * R + r;
        const unsigned ii = (unsigned)in_inds[base];
        const unsigned oi = (unsigned)out_inds[base];
        const float* xr = x_data + (size_t)(ii * (unsigned)C_IN);
        const float* wk = weights + (size_t)k * C_IN * C_OUT;
        float acc = 0.f;
#pragma unroll
        for (int ci = 0; ci < C_IN; ++ci) acc += xr[ci] * wk[ci * C_OUT + c];
        atomic_add_f32(out + (size_t)(oi * (unsigned)C_OUT + (unsigned)c), acc);
    }
}

extern "C" void kernel_launch(void* const* d_in, const int* in_sizes, int n_in,
                              void* d_out, int out_size, void* d_ws, size_t ws_size,
                              hipStream_t stream) {
    const float* x_data   = (const float*)d_in[0];
    const float* weights  = (const float*)d_in[1];
    const float* bias     = (const float*)d_in[2];
    const int*   in_inds  = (const int*)d_in[3];
    const int*   out_inds = (const int*)d_in[4];
    float*       out      = (float*)d_out;

    const int R = in_sizes[3] / K3;                       // rows per kernel offset
    const long long n_rows = (long long)out_size / C_OUT; // output rows

    // 1) out = bias broadcast
    const long long total4 = n_rows * (C_OUT / 4);
    const int threads = 256;
    const unsigned blocks = (unsigned)((total4 + threads - 1) / threads);
    init_bias_kernel<<<blocks, threads, 0, stream>>>(out, bias, total4);

    // 2) gather -> WMMA -> scatter-add over complete 16-row groups.
    //    Persistent waves: cap x-blocks so each wave loops ~15 groups,
    //    amortizing weight staging + B-fragment setup.
    const int full_groups = R / 16;
    if (full_groups > 0) {
        int bx = (full_groups + 7) / 8;
        if (bx > 512) bx = 512;
        dim3 grid(bx, K3);
        sparse_conv_gather_mm_scatter<<<grid, 256, 0, stream>>>(
            x_data, weights, in_inds, out_inds, out, R);
    }

    // 3) tail rows (R % 16) — host-known size, so this is capture-deterministic
    if (R % 16 != 0) {
        sparse_conv_tail<<<K3, 32, 0, stream>>>(
            x_data, weights, in_inds, out_inds, out, R, full_groups * 16);
    }
}